// RecurrentCrossLinearAttention_2688649527590
// MI455X (gfx1250) — compile-verified
//
#include <hip/hip_runtime.h>

typedef __attribute__((ext_vector_type(2))) float        v2f;
typedef __attribute__((ext_vector_type(4))) float        f4;
typedef __attribute__((ext_vector_type(8))) float        v8f;
typedef __attribute__((ext_vector_type(4))) unsigned int v4u;
typedef __attribute__((ext_vector_type(4))) int          v4i;
typedef __attribute__((ext_vector_type(8))) int          v8i;

#define N_    16
#define S_    4096
#define H_    16
#define D_    64
#define M_    64
#define CHUNK 64
#define LSTR  80   // padded LDS row stride (floats) == TDM pad: 16 DW per 64 DW

#if defined(__has_builtin)
#if __has_builtin(__builtin_amdgcn_tensor_load_to_lds) && \
    __has_builtin(__builtin_amdgcn_s_wait_tensorcnt)
#define HAVE_TDM 1
#endif
#endif

// out layout (flat, reference return order):
//   V_out  [N,H,M]   : offset 0,        16384 floats
//   S_state[N,H,M,D] : offset 16384,    1048576 floats
//   Z      [N,H,D]   : offset 1064960,  16384 floats

#ifdef HAVE_TDM
// DMA one 64x64 f32 tile (row stride 1024 elems) from global -> LDS with
// 16-DWORD padding every 64 DWORDs (matches LSTR=80).
__device__ __forceinline__ void tdm_load_tile(const float* gsrc, unsigned lds_off) {
  unsigned long long ga = (unsigned long long)(uintptr_t)gsrc;
  v4u g0;
  g0[0] = 1u;                                            // count=1 (valid), user D#
  g0[1] = lds_off;                                       // LDS byte address
  g0[2] = (unsigned)(ga & 0xffffffffu);                  // global_addr[31:0]
  g0[3] = (unsigned)((ga >> 32) & 0x01ffffffu) | (2u << 30); // addr[56:32] | type=2
  v8i g1;
  g1[0] = (int)((2u << 16)            // data_size = 4B
              | (1u << 20)            // pad_enable
              | (5u << 22)            // pad_interval: 64 DWORDs
              | (15u << 25));         // pad_amount: 16 DWORDs
  g1[1] = (int)(64u << 16);           // tensor_dim0 = 64 (bits 79:48 lo16)
  g1[2] = (int)((unsigned)S_ << 16);  // tensor_dim0 hi | tensor_dim1 lo16 = 4096
  g1[3] = (int)(64u << 16);           // tensor_dim1 hi | tile_dim0 = 64
  g1[4] = 64;                         // tile_dim1 = 64, tile_dim2 = 0 (2D)
  g1[5] = (int)(H_ * M_);             // tensor_dim0_stride = 1024 elems (lo32)
  g1[6] = 0;                          // stride0 hi | stride1 lo
  g1[7] = 0;                          // stride1 hi
  v4i gz4  = {0, 0, 0, 0};            // groups 2/3 unused for 2D tile
  v8i gz8  = {0, 0, 0, 0, 0, 0, 0, 0};
  // amdgpu-toolchain (clang-23) 6-arg form
  __builtin_amdgcn_tensor_load_to_lds(g0, g1, gz4, gz4, gz8, 0);
}
#endif

__global__ __launch_bounds__(256) void lin_attn_kernel(
    const float* __restrict__ query,
    const float* __restrict__ keys,
    const float* __restrict__ values,
    const float* __restrict__ key_mask,
    float* __restrict__ out)
{
  __shared__ __align__(16) float Ks[CHUNK * LSTR];
  __shared__ __align__(16) float Vs[2][CHUNK * LSTR];
  __shared__ float qs[64];
  __shared__ float zsum[64];
  __shared__ float vout_s[64];
  __shared__ float red[64];
  __shared__ float qz_s;

  const int nh   = blockIdx.x;      // (n*H + h)
  const int n    = nh / H_;
  const int h    = nh % H_;
  const int t    = threadIdx.x;     // 0..255
  const int lane = t & 31;          // wave32
  const int wave = t >> 5;          // 0..7
  const int half = lane >> 4;       // lanes 16-31 hold k+2 / m+8
  const int lcol = lane & 15;

  // K loader mapping (b128): row lr + 16*pass, 4 consecutive cols at lc4
  const int lr  = t >> 4;           // 0..15
  const int lc4 = (t & 15) * 4;     // 0,4,...,60

  if (t < 64) {
    float x = query[nh * 64 + t];
    qs[t] = (x > 0.f) ? (x + 1.f) : __expf(x);   // elu(x)+1
    vout_s[t] = 0.f;
    zsum[t]   = 0.f;
  }

  // wave -> two adjacent d-tiles sharing one m-tile
  const int m0  = (wave >> 1) * 16;
  const int d0a = (wave & 1) * 32;
  const int d0b = d0a + 16;

  v8f acc0 = {0.f,0.f,0.f,0.f,0.f,0.f,0.f,0.f};
  v8f acc1 = {0.f,0.f,0.f,0.f,0.f,0.f,0.f,0.f};
  float zp0 = 0.f, zp1 = 0.f, zp2 = 0.f, zp3 = 0.f;

  const float* kbase = keys   + ((size_t)n * S_ * H_ + h) * D_;
  const float* vbase = values + ((size_t)n * S_ * H_ + h) * M_;
  const float* mbase = key_mask + (size_t)n * S_;

#ifdef HAVE_TDM
  if (wave == 0) {
    tdm_load_tile(vbase, (unsigned)(uintptr_t)&Vs[0][0]);  // prologue: chunk 0
  }
#endif

  for (int ci = 0; ci < S_ / CHUNK; ci++) {
    const int s0 = ci * CHUNK;
    const int vb = ci & 1;
    __syncthreads();   // compute of ci-1 done: Ks and Vs[(ci+1)&1] reusable

    // ---- K chunk: b128 loads, ELU+1, mask, Z partials ----
    #pragma unroll
    for (int p = 0; p < 4; p++) {
      const int rr = lr + 16 * p;
      const int s  = s0 + rr;
      f4 kv = *(const f4*)(kbase + (size_t)s * (H_ * D_) + lc4);
      const float mk = mbase[s];
      float k0 = ((kv[0] > 0.f) ? (kv[0] + 1.f) : __expf(kv[0])) * mk;
      float k1 = ((kv[1] > 0.f) ? (kv[1] + 1.f) : __expf(kv[1])) * mk;
      float k2 = ((kv[2] > 0.f) ? (kv[2] + 1.f) : __expf(kv[2])) * mk;
      float k3 = ((kv[3] > 0.f) ? (kv[3] + 1.f) : __expf(kv[3])) * mk;
      zp0 += k0; zp1 += k1; zp2 += k2; zp3 += k3;
      f4 kf; kf[0] = k0; kf[1] = k1; kf[2] = k2; kf[3] = k3;
      *(f4*)(Ks + rr * LSTR + lc4) = kf;
    }
    if (s0 + CHUNK < S_) {  // gfx1250 global_prefetch for next K chunk
      __builtin_prefetch(kbase + (size_t)(s0 + CHUNK + lr) * (H_ * D_) + lc4, 0, 1);
    }

#ifdef HAVE_TDM
    if (wave == 0) {
      if (ci + 1 < S_ / CHUNK) {
        tdm_load_tile(vbase + (size_t)(s0 + CHUNK) * (H_ * M_),
                      (unsigned)(uintptr_t)&Vs[vb ^ 1][0]);
        __builtin_amdgcn_s_wait_tensorcnt(1);   // chunk ci landed, ci+1 in flight
      } else {
        __builtin_amdgcn_s_wait_tensorcnt(0);
      }
    }
#else
    #pragma unroll
    for (int p = 0; p < 4; p++) {
      const int rr = lr + 16 * p;
      *(f4*)(Vs[vb] + rr * LSTR + lc4) =
          *(const f4*)(vbase + (size_t)(s0 + rr) * (H_ * M_) + lc4);
    }
#endif
    __syncthreads();

    // ---- 16 k-steps x 2 WMMA: S_state += V^T x K ----
    #pragma unroll
    for (int kk = 0; kk < CHUNK; kk += 4) {
      const int ra = kk + 2 * half;       // this lane-half holds k = ra, ra+1
      v2f a, b0, b1;
      a[0]  = Vs[vb][(ra + 0) * LSTR + m0  + lcol];  // A[m][k] = V[s][m]
      a[1]  = Vs[vb][(ra + 1) * LSTR + m0  + lcol];
      b0[0] = Ks[(ra + 0) * LSTR + d0a + lcol];      // B[k][d] = K[s][d]
      b0[1] = Ks[(ra + 1) * LSTR + d0a + lcol];
      b1[0] = Ks[(ra + 0) * LSTR + d0b + lcol];
      b1[1] = Ks[(ra + 1) * LSTR + d0b + lcol];
      acc0 = __builtin_amdgcn_wmma_f32_16x16x4_f32(false, a, false, b0,
                                                   (short)0, acc0, false, false);
      acc1 = __builtin_amdgcn_wmma_f32_16x16x4_f32(false, a, false, b1,
                                                   (short)0, acc1, false, false);
    }
  }

  // ---- Z = column-sum of K over S (ds_add_f32 reduction) ----
  atomicAdd(&zsum[lc4 + 0], zp0);
  atomicAdd(&zsum[lc4 + 1], zp1);
  atomicAdd(&zsum[lc4 + 2], zp2);
  atomicAdd(&zsum[lc4 + 3], zp3);
  __syncthreads();
  if (t < 64) {
    const float z = zsum[t];
    out[1064960 + nh * 64 + t] = z;          // Z output
    red[t] = z * qs[t];
  }
  __syncthreads();
  if (t == 0) {
    float s = 0.f;
    #pragma unroll
    for (int i = 0; i < 64; i++) s += red[i];
    qz_s = 1.0f / (s + 1e-6f);               // QZ = 1/(Q.Z + eps)
  }

  // ---- write S_state, accumulate V_out partials ----
  float* sbase = out + 16384 + (size_t)nh * (M_ * D_);
  #pragma unroll
  for (int r = 0; r < 8; r++) {
    const int m = m0 + r + 8 * half;
    const float c0 = acc0[r];
    const float c1 = acc1[r];
    sbase[m * D_ + d0a + lcol] = c0;
    sbase[m * D_ + d0b + lcol] = c1;
    atomicAdd(&vout_s[m], qs[d0a + lcol] * c0 + qs[d0b + lcol] * c1);
  }
  __syncthreads();
  if (t < 64) {
    out[nh * 64 + t] = vout_s[t] * qz_s;     // V_out
  }
}

extern "C" void kernel_launch(void* const* d_in, const int* in_sizes, int n_in,
                              void* d_out, int out_size, void* d_ws, size_t ws_size,
                              hipStream_t stream) {
  const float* query    = (const float*)d_in[0];
  const float* keys     = (const float*)d_in[1];
  const float* values   = (const float*)d_in[2];
  const float* key_mask = (const float*)d_in[3];
  float* out = (float*)d_out;
  (void)in_sizes; (void)n_in; (void)out_size; (void)d_ws; (void)ws_size;

  dim3 grid(N_ * H_);    // one workgroup per (n,h)
  dim3 block(256);       // 8 wave32s
  lin_attn_kernel<<<grid, block, 0, stream>>>(query, keys, values, key_mask, out);
}